// InteractionBlock_83665962926266
// MI455X (gfx1250) — compile-verified
//
#include <hip/hip_runtime.h>

typedef __attribute__((ext_vector_type(2))) float v2f;
typedef __attribute__((ext_vector_type(8))) float v8f;

#define EMB 128
#define AINT 64
#define NRBF 6
#define NSBF 42
#define SBF_KPAD 44
#define LDSP 132   // LDS row pitch (floats): (4m+k)%64 distinct banks across lanes

__device__ __forceinline__ float swishf(float x) {
    return x / (1.0f + __expf(-x));
}

// One 16x16 fp32 output tile via V_WMMA_F32_16X16X4_F32, K unrolled.
// A: row-major [16,K] (global or LDS), W: row-major [K,N] (global).
// arow = lane&15, col = absolute output column for this lane, ko = (lane>>4)*2.
template <int K>
__device__ __forceinline__ v8f gemm_tile(const float* __restrict__ A, int lda,
                                         const float* __restrict__ W, int ldw,
                                         int arow, int col, int ko) {
    v8f acc = {};
#pragma unroll
    for (int k = 0; k < K; k += 4) {
        v2f a, b;
        a.x = A[arow * lda + k + ko + 0];
        a.y = A[arow * lda + k + ko + 1];
        b.x = W[(k + ko + 0) * ldw + col];
        b.y = W[(k + ko + 1) * ldw + col];
        acc = __builtin_amdgcn_wmma_f32_16x16x4_f32(false, a, false, b,
                                                    (short)0, acc, false, false);
    }
    return acc;
}

// ---------------- K0: fold tiny basis MLPs into single matrices ----------------
// W_rbf[6][128] = rbf1 @ rbf2 ; W_sbf[44][64] = sbf1 @ sbf2 (rows 42..43 zero pad)
__global__ void prep_kernel(const float* __restrict__ rbf1, const float* __restrict__ rbf2,
                            const float* __restrict__ sbf1, const float* __restrict__ sbf2,
                            float* __restrict__ W_rbf, float* __restrict__ W_sbf) {
    int idx = threadIdx.x + blockIdx.x * blockDim.x;
    int stride = blockDim.x * gridDim.x;
    for (int i = idx; i < NRBF * EMB; i += stride) {
        int r = i / EMB, c = i % EMB;
        float s = 0.f;
#pragma unroll
        for (int j = 0; j < 8; ++j) s += rbf1[r * 8 + j] * rbf2[j * EMB + c];
        W_rbf[i] = s;
    }
    for (int i = idx; i < SBF_KPAD * AINT; i += stride) {
        int r = i / AINT, c = i % AINT;
        float s = 0.f;
        if (r < NSBF) {
#pragma unroll
            for (int j = 0; j < 8; ++j) s += sbf1[r * 8 + j] * sbf2[j * AINT + c];
        }
        W_sbf[i] = s;
    }
}

// ---------------- K1: zero the segment-sum accumulator ----------------
__global__ void zero_kernel(float* __restrict__ p, size_t n) {
    size_t i = (size_t)blockIdx.x * blockDim.x + threadIdx.x;
    size_t stride = (size_t)gridDim.x * blockDim.x;
    for (; i < n; i += stride) p[i] = 0.0f;
}

// ---------------- K2: edge front ----------------
// m_ang2 = swish( (swish(m_input@Wkj + bkj) * (rbf@W_rbf)) @ down )   -> [E, 64]
#define NW_E 8
__global__ __launch_bounds__(NW_E * 32) void edge_front_kernel(
    const float* __restrict__ m_input, const float* __restrict__ rbf,
    const float* __restrict__ Wkj, const float* __restrict__ bkj,
    const float* __restrict__ W_rbf, const float* __restrict__ down,
    float* __restrict__ m_ang2, int E) {
    __shared__ float smem[NW_E][16 * LDSP];
    int lane = threadIdx.x & 31;
    int wave = threadIdx.x >> 5;
    int ebase = (blockIdx.x * NW_E + wave) * 16;
    if (ebase >= E) return;
    int arow = lane & 15;
    int half = lane >> 4;
    int ko = half * 2;
    float* cur = smem[wave];
    const float* Ain = m_input + (size_t)ebase * EMB;

    __builtin_prefetch(Wkj + (size_t)lane * 512, 0, 1);  // global_prefetch_b8

    // h = swish(m_input @ Wkj + bkj) * rbf_p  -> cur (LDS)
#pragma unroll 1
    for (int n = 0; n < 8; ++n) {
        int col = n * 16 + arow;
        v8f acc = gemm_tile<EMB>(Ain, EMB, Wkj, EMB, arow, col, ko);
        float bias = bkj[col];
        float wr[NRBF];
#pragma unroll
        for (int j = 0; j < NRBF; ++j) wr[j] = W_rbf[j * EMB + col];
#pragma unroll
        for (int r = 0; r < 8; ++r) {
            int m = r + 8 * half;
            float rp = 0.f;
#pragma unroll
            for (int j = 0; j < NRBF; ++j) rp += rbf[(size_t)(ebase + m) * NRBF + j] * wr[j];
            cur[m * LDSP + col] = swishf(acc[r] + bias) * rp;
        }
    }
    __syncthreads();
    // m_ang2 = swish(cur @ down)
#pragma unroll 1
    for (int n = 0; n < 4; ++n) {
        int col = n * 16 + arow;
        v8f acc = gemm_tile<EMB>(cur, LDSP, down, AINT, arow, col, ko);
#pragma unroll
        for (int r = 0; r < 8; ++r) {
            int m = r + 8 * half;
            m_ang2[(size_t)(ebase + m) * AINT + col] = swishf(acc[r]);
        }
    }
}

// ---------------- K3: triplets ----------------
// agg[ji] += m_ang2[kj] * (sbf @ W_sbf)   (segment sum via L2 atomics)
#define NW_T 8
__global__ __launch_bounds__(NW_T * 32) void triplet_kernel(
    const float* __restrict__ sbf, const int* __restrict__ red_ji,
    const int* __restrict__ exp_kj, const float* __restrict__ W_sbf,
    const float* __restrict__ m_ang2, float* __restrict__ agg, int T) {
    __shared__ float ws[SBF_KPAD * AINT];
    for (int i = threadIdx.x; i < SBF_KPAD * AINT; i += blockDim.x) ws[i] = W_sbf[i];
    __syncthreads();
    int lane = threadIdx.x & 31;
    int wave = threadIdx.x >> 5;
    int tbase = (blockIdx.x * NW_T + wave) * 16;
    if (tbase >= T) return;
    int arow = lane & 15;
    int half = lane >> 4;
    int ko = half * 2;
    const float* Arow = sbf + (size_t)(tbase + arow) * NSBF;

#pragma unroll 1
    for (int n = 0; n < 4; ++n) {
        int col = n * 16 + arow;
        v8f acc = {};
#pragma unroll
        for (int k = 0; k < SBF_KPAD; k += 4) {
            int k0 = k + ko, k1 = k + ko + 1;
            v2f a, b;
            a.x = (k0 < NSBF) ? Arow[k0] : 0.0f;
            a.y = (k1 < NSBF) ? Arow[k1] : 0.0f;
            b.x = ws[k0 * AINT + col];   // padded rows are zero
            b.y = ws[k1 * AINT + col];
            acc = __builtin_amdgcn_wmma_f32_16x16x4_f32(false, a, false, b,
                                                        (short)0, acc, false, false);
        }
#pragma unroll
        for (int r = 0; r < 8; ++r) {
            int t = tbase + r + 8 * half;
            int kj = exp_kj[t];
            int ji = red_ji[t];
            float v = acc[r] * m_ang2[(size_t)kj * AINT + col];
            atomicAdd(&agg[(size_t)ji * AINT + col], v);   // global_atomic_add_f32, L2-resident
        }
    }
}

// ---------------- K4: edge back (fully fused GEMM chain) ----------------
#define NW_B 4
__device__ __forceinline__ void res_layer(float* cur, float* tmp,
                                          const float* __restrict__ w1, const float* __restrict__ b1,
                                          const float* __restrict__ w2, const float* __restrict__ b2,
                                          int arow, int half, int ko) {
#pragma unroll 1
    for (int n = 0; n < 8; ++n) {
        int col = n * 16 + arow;
        v8f acc = gemm_tile<EMB>(cur, LDSP, w1, EMB, arow, col, ko);
        float bias = b1[col];
#pragma unroll
        for (int r = 0; r < 8; ++r) tmp[(r + 8 * half) * LDSP + col] = swishf(acc[r] + bias);
    }
    __syncthreads();
#pragma unroll 1
    for (int n = 0; n < 8; ++n) {
        int col = n * 16 + arow;
        v8f acc = gemm_tile<EMB>(tmp, LDSP, w2, EMB, arow, col, ko);
        float bias = b2[col];
#pragma unroll
        for (int r = 0; r < 8; ++r) cur[(r + 8 * half) * LDSP + col] += swishf(acc[r] + bias);
    }
    __syncthreads();
}

__global__ __launch_bounds__(NW_B * 32) void edge_back_kernel(
    const float* __restrict__ m_input, const float* __restrict__ agg,
    const float* __restrict__ up,
    const float* __restrict__ Wji, const float* __restrict__ bji,
    const float* __restrict__ rb_w1, const float* __restrict__ rb_b1,
    const float* __restrict__ rb_w2, const float* __restrict__ rb_b2,
    const float* __restrict__ fw, const float* __restrict__ fb,
    const float* __restrict__ ra0_w1, const float* __restrict__ ra0_b1,
    const float* __restrict__ ra0_w2, const float* __restrict__ ra0_b2,
    const float* __restrict__ ra1_w1, const float* __restrict__ ra1_b1,
    const float* __restrict__ ra1_w2, const float* __restrict__ ra1_b2,
    float* __restrict__ out, int E) {
    __shared__ float smem[NW_B][2][16 * LDSP];
    int lane = threadIdx.x & 31;
    int wave = threadIdx.x >> 5;
    int ebase = (blockIdx.x * NW_B + wave) * 16;
    if (ebase >= E) return;
    int arow = lane & 15;
    int half = lane >> 4;
    int ko = half * 2;
    float* cur = smem[wave][0];
    float* tmp = smem[wave][1];
    const float* Ain = m_input + (size_t)ebase * EMB;
    const float* Agg = agg + (size_t)ebase * AINT;

    // m = swish(m_input@Wji + bji) + swish(agg@up)
#pragma unroll 1
    for (int n = 0; n < 8; ++n) {
        int col = n * 16 + arow;
        v8f a1 = gemm_tile<AINT>(Agg, AINT, up, EMB, arow, col, ko);
        v8f a2 = gemm_tile<EMB>(Ain, EMB, Wji, EMB, arow, col, ko);
        float bias = bji[col];
#pragma unroll
        for (int r = 0; r < 8; ++r)
            cur[(r + 8 * half) * LDSP + col] = swishf(a2[r] + bias) + swishf(a1[r]);
    }
    __syncthreads();

    // 1 residual layer before skip
    res_layer(cur, tmp, rb_w1, rb_b1, rb_w2, rb_b2, arow, half, ko);

    // m = swish(m@final + fb) + m_input   (result lands in tmp)
#pragma unroll 1
    for (int n = 0; n < 8; ++n) {
        int col = n * 16 + arow;
        v8f acc = gemm_tile<EMB>(cur, LDSP, fw, EMB, arow, col, ko);
        float bias = fb[col];
#pragma unroll
        for (int r = 0; r < 8; ++r) {
            int m = r + 8 * half;
            tmp[m * LDSP + col] = swishf(acc[r] + bias) + Ain[m * EMB + col];
        }
    }
    __syncthreads();

    // 2 residual layers after skip (activation now in tmp; cur is scratch)
    res_layer(tmp, cur, ra0_w1, ra0_b1, ra0_w2, ra0_b2, arow, half, ko);
    res_layer(tmp, cur, ra1_w1, ra1_b1, ra1_w2, ra1_b2, arow, half, ko);

    // writeback
    for (int i = lane; i < 16 * EMB; i += 32) {
        int m = i >> 7, c = i & 127;
        out[(size_t)(ebase + m) * EMB + c] = tmp[m * LDSP + c];
    }
}

extern "C" void kernel_launch(void* const* d_in, const int* in_sizes, int n_in,
                              void* d_out, int out_size, void* d_ws, size_t ws_size,
                              hipStream_t stream) {
    // setup_inputs() dict order, params flattened in insertion order
    const float* m_input = (const float*)d_in[0];
    const float* rbf     = (const float*)d_in[1];
    const float* sbf     = (const float*)d_in[2];
    const int*   red_ji  = (const int*)d_in[3];
    const int*   exp_kj  = (const int*)d_in[4];
    const float* rbf1    = (const float*)d_in[5];
    const float* rbf2    = (const float*)d_in[6];
    const float* sbf1    = (const float*)d_in[7];
    const float* sbf2    = (const float*)d_in[8];
    const float* Wkj     = (const float*)d_in[9];
    const float* bkj     = (const float*)d_in[10];
    const float* down    = (const float*)d_in[11];
    const float* up      = (const float*)d_in[12];
    const float* Wji     = (const float*)d_in[13];
    const float* bji     = (const float*)d_in[14];
    const float* fw      = (const float*)d_in[15];
    const float* fb      = (const float*)d_in[16];
    const float* rb_w1   = (const float*)d_in[17];
    const float* rb_b1   = (const float*)d_in[18];
    const float* rb_w2   = (const float*)d_in[19];
    const float* rb_b2   = (const float*)d_in[20];
    const float* ra0_w1  = (const float*)d_in[21];
    const float* ra0_b1  = (const float*)d_in[22];
    const float* ra0_w2  = (const float*)d_in[23];
    const float* ra0_b2  = (const float*)d_in[24];
    const float* ra1_w1  = (const float*)d_in[25];
    const float* ra1_b1  = (const float*)d_in[26];
    const float* ra1_w2  = (const float*)d_in[27];
    const float* ra1_b2  = (const float*)d_in[28];

    const int E = in_sizes[0] / EMB;
    const int T = in_sizes[2] / NSBF;

    float* ws      = (float*)d_ws;
    float* W_rbf   = ws;                                  // 6*128
    float* W_sbf   = ws + 768;                            // 44*64
    float* m_ang2  = ws + 4096;                           // E*64
    float* agg     = m_ang2 + (size_t)E * AINT;           // E*64

    prep_kernel<<<4, 256, 0, stream>>>(rbf1, rbf2, sbf1, sbf2, W_rbf, W_sbf);
    zero_kernel<<<1024, 256, 0, stream>>>(agg, (size_t)E * AINT);
    edge_front_kernel<<<E / (16 * NW_E), NW_E * 32, 0, stream>>>(
        m_input, rbf, Wkj, bkj, W_rbf, down, m_ang2, E);
    triplet_kernel<<<T / (16 * NW_T), NW_T * 32, 0, stream>>>(
        sbf, red_ji, exp_kj, W_sbf, m_ang2, agg, T);
    edge_back_kernel<<<E / (16 * NW_B), NW_B * 32, 0, stream>>>(
        m_input, agg, up, Wji, bji, rb_w1, rb_b1, rb_w2, rb_b2, fw, fb,
        ra0_w1, ra0_b1, ra0_w2, ra0_b2, ra1_w1, ra1_b1, ra1_w2, ra1_b2,
        (float*)d_out, E);
}